// CosformerAttention_41738492182531
// MI455X (gfx1250) — compile-verified
//
#include <hip/hip_runtime.h>
#include <stdint.h>

typedef float v2f __attribute__((ext_vector_type(2)));
typedef float v8f __attribute__((ext_vector_type(8)));

#define L_SEQ   1024
#define NBATCH  2
#define E_DIM   512
#define NHEADS  8
#define DHEAD   64
#define NH      16            // NBATCH * NHEADS
#define CHUNK   16
#define NCHUNK  (L_SEQ / CHUNK)
#define EPSV    1e-6f
#define PI_HALF 1.57079632679489662f
#define XPAD    516           // padded LDS row stride for 512-wide A panel
#define TPAD    68            // padded LDS row stride for 64-wide q/k/v tiles

// D = A(16x4 f32) * B(4x16 f32) + C(16x16 f32)
static __device__ __forceinline__ v8f wmma4(v2f a, v2f b, v8f c) {
  return __builtin_amdgcn_wmma_f32_16x16x4_f32(false, a, false, b,
                                               (short)0, c, false, false);
}

// 16-byte async global -> LDS copy (ASYNCcnt-tracked, bypasses VGPRs)
static __device__ __forceinline__ void async_copy16(uint32_t lds_off,
                                                    const void* gptr) {
  asm volatile("global_load_async_to_lds_b128 %0, %1, off"
               :: "v"(lds_off), "v"(gptr) : "memory");
}
static __device__ __forceinline__ void wait_async0() {
  asm volatile("s_wait_asynccnt 0x0" ::: "memory");
}

// ---------------------------------------------------------------------------
// Dense projection GEMM: Y[m][n] = act( sum_k X[m][k]*W[n][k] + bias[n] )
// M = L*N = 2048, K = 512, Nout = 512.  One 16x16 C-tile per wave.
// A panel (16 x 512) staged once into LDS via async copies, shared by 4 waves.
// ---------------------------------------------------------------------------
__global__ __launch_bounds__(128) void cosformer_proj_gemm(
    const float* __restrict__ X, const float* __restrict__ W,
    const float* __restrict__ Bi, float* __restrict__ Y, int relu) {
  __shared__ __align__(16) float sX[16][XPAD];   // ~32.3 KB

  const int tid  = threadIdx.x;
  const int lane = tid & 31;
  const int wave = tid >> 5;
  const int hi   = lane >> 4;          // K-pair select within fragment
  const int lcol = lane & 15;          // A: M row / B: N col
  const int mBase = blockIdx.y << 4;
  const int nBase = (blockIdx.x << 6) + (wave << 4);

  // ---- async-stage A panel: 16 rows x 512 floats = 2048 x 16B transfers ----
  for (int t = tid; t < 16 * 128; t += 128) {
    const int row = t >> 7;
    const int c4  = (t & 127) << 2;    // float index of 16B unit
    async_copy16((uint32_t)(uintptr_t)&sX[row][c4],
                 X + (size_t)(mBase + row) * E_DIM + c4);
  }
  wait_async0();
  __syncthreads();

  const float* wrow = W + (size_t)(nBase + lcol) * E_DIM + 2 * hi;

  v8f acc = {0.f, 0.f, 0.f, 0.f, 0.f, 0.f, 0.f, 0.f};
  #pragma unroll 8
  for (int k0 = 0; k0 < E_DIM; k0 += 4) {
    if ((k0 & 63) == 0) __builtin_prefetch(wrow + k0 + 128, 0, 1);
    v2f a = *(const v2f*)(&sX[lcol][k0 + 2 * hi]);
    v2f b = *(const v2f*)(wrow + k0);
    acc = wmma4(a, b, acc);
  }

  const float bn = Bi[nBase + lcol];
  #pragma unroll
  for (int r = 0; r < 8; ++r) {
    float v = acc[r] + bn;
    if (relu) v = fmaxf(v, 0.0f);
    Y[(size_t)(mBase + r + 8 * hi) * E_DIM + nBase + lcol] = v;
  }
}

// ---------------------------------------------------------------------------
// Chunked causal linear-attention scan. One block per head, 5 waves.
// Wave w owns columns [16w, 16w+16) of the augmented V' (cols 0..63 = v,
// col 64 = ones -> denominator). State S (128 x 80) lives in LDS.
// Chunk q/k/v tiles are async-staged into LDS each iteration.
// ---------------------------------------------------------------------------
__global__ __launch_bounds__(160) void cosformer_scan(
    const float* __restrict__ qbuf, const float* __restrict__ kbuf,
    const float* __restrict__ vbuf, float* __restrict__ attn) {
  __shared__ __align__(16) float sS[5][128][16];   // per-wave state (40 KB)
  __shared__ __align__(16) float sA[5][16][17];    // masked-A staging (padded)
  __shared__ __align__(16) float sQ[16][TPAD];     // chunk q tile
  __shared__ __align__(16) float sK[16][TPAD];     // chunk k tile
  __shared__ __align__(16) float sV[16][TPAD];     // chunk v tile
  __shared__ float sSin[16], sCos[16];             // chunk sin/cos table
  __shared__ float sDen[2][16];                    // denom (double buffered)

  const int tid  = threadIdx.x;
  const int lane = tid & 31;
  const int wave = tid >> 5;
  const int hi   = lane >> 4;
  const int lcol = lane & 15;
  const int head = blockIdx.x;
  const int nIdx = head >> 3;          // batch index
  const int fb   = (head & 7) * DHEAD; // feature base within E

  for (int i = tid; i < 5 * 128 * 16; i += blockDim.x)
    (&sS[0][0][0])[i] = 0.0f;
  __syncthreads();

  const float thScale = PI_HALF / (float)L_SEQ;

  for (int ch = 0; ch < NCHUNK; ++ch) {
    const int base = ch * CHUNK;

    // ---- async-stage chunk tiles: 3 tensors x 16 rows x 16 x 16B ----
    for (int t = tid; t < 768; t += 160) {
      const int tensor = t >> 8;
      const int rem    = t & 255;
      const int row    = rem >> 4;
      const int c4     = (rem & 15) << 2;
      const size_t goff =
          (size_t)((base + row) * NBATCH + nIdx) * E_DIM + fb + c4;
      const float* g;
      uint32_t l;
      if (tensor == 0)      { g = qbuf + goff; l = (uint32_t)(uintptr_t)&sQ[row][c4]; }
      else if (tensor == 1) { g = kbuf + goff; l = (uint32_t)(uintptr_t)&sK[row][c4]; }
      else                  { g = vbuf + goff; l = (uint32_t)(uintptr_t)&sV[row][c4]; }
      async_copy16(l, g);
    }
    if (tid < 16) {
      const float th = thScale * (float)(base + tid + 1);
      sSin[tid] = __sinf(th);
      sCos[tid] = __cosf(th);
    }
    wait_async0();
    __syncthreads();

    const float sl = sSin[lcol];
    const float cl = sCos[lcol];

    // ---- (1) inter-chunk: out = Q~ @ S_prev  (K = 128) ----
    v8f outAcc = {0.f, 0.f, 0.f, 0.f, 0.f, 0.f, 0.f, 0.f};
    #pragma unroll 4
    for (int ks = 0; ks < 32; ++ks) {
      const int j = 4 * ks + 2 * hi;
      const float sc = (j < 64) ? sl : cl;
      v2f a = *(const v2f*)(&sQ[lcol][j & 63]);
      a.x *= sc; a.y *= sc;
      v2f b;
      b.x = sS[wave][j][lcol];
      b.y = sS[wave][j + 1][lcol];
      outAcc = wmma4(a, b, outAcc);
    }

    // ---- (2) A = Q~ K~^T (16x16, K = 128), then causal mask ----
    v8f aAcc = {0.f, 0.f, 0.f, 0.f, 0.f, 0.f, 0.f, 0.f};
    #pragma unroll 4
    for (int ks = 0; ks < 32; ++ks) {
      const int j = 4 * ks + 2 * hi;
      const float sc = (j < 64) ? sl : cl;
      v2f a = *(const v2f*)(&sQ[lcol][j & 63]);
      v2f b = *(const v2f*)(&sK[lcol][j & 63]);
      a.x *= sc; a.y *= sc;
      b.x *= sc; b.y *= sc;
      aAcc = wmma4(a, b, aAcc);
    }
    #pragma unroll
    for (int r = 0; r < 8; ++r) {
      const int row = r + 8 * hi;                  // query index in chunk
      sA[wave][row][lcol] = (lcol <= row) ? aAcc[r] : 0.0f;
    }
    // wave-synchronous LDS transpose: wait our own DS stores
    asm volatile("s_wait_dscnt 0x0" ::: "memory");

    // ---- (3) intra-chunk: out += mask(A) @ V'  (K = 16) ----
    #pragma unroll
    for (int ks = 0; ks < 4; ++ks) {
      const int t = 4 * ks + 2 * hi;
      v2f a;
      a.x = sA[wave][lcol][t];
      a.y = sA[wave][lcol][t + 1];
      v2f b;
      if (wave < 4) {
        b.x = sV[t][wave * 16 + lcol];
        b.y = sV[t + 1][wave * 16 + lcol];
      } else {
        const float one = (lcol == 0) ? 1.0f : 0.0f;
        b.x = one; b.y = one;
      }
      outAcc = wmma4(a, b, outAcc);
    }

    // ---- (4) state update: S += K~^T @ V'  (128x16, K = 16) ----
    #pragma unroll 2
    for (int mt = 0; mt < 8; ++mt) {
      v8f sAcc = {0.f, 0.f, 0.f, 0.f, 0.f, 0.f, 0.f, 0.f};
      const int j  = 16 * mt + lcol;    // feature row of S
      const int jm = j & 63;
      const bool lo = (j < 64);
      #pragma unroll
      for (int ks = 0; ks < 4; ++ks) {
        const int t = 4 * ks + 2 * hi;
        const float sc0 = lo ? sSin[t] : sCos[t];
        const float sc1 = lo ? sSin[t + 1] : sCos[t + 1];
        v2f a;
        a.x = sc0 * sK[t][jm];
        a.y = sc1 * sK[t + 1][jm];
        v2f b;
        if (wave < 4) {
          b.x = sV[t][wave * 16 + lcol];
          b.y = sV[t + 1][wave * 16 + lcol];
        } else {
          const float one = (lcol == 0) ? 1.0f : 0.0f;
          b.x = one; b.y = one;
        }
        sAcc = wmma4(a, b, sAcc);
      }
      #pragma unroll
      for (int r = 0; r < 8; ++r)
        sS[wave][16 * mt + r + 8 * hi][lcol] += sAcc[r];
    }

    // ---- (5) denominator broadcast + normalized output store ----
    if (wave == 4 && lcol == 0) {
      #pragma unroll
      for (int r = 0; r < 8; ++r) sDen[ch & 1][r + 8 * hi] = outAcc[r];
    }
    __syncthreads();
    if (wave < 4) {
      const int colg = fb + wave * 16 + lcol;
      #pragma unroll
      for (int r = 0; r < 8; ++r) {
        const int row = r + 8 * hi;
        const float den = fmaxf(sDen[ch & 1][row], EPSV);
        attn[(size_t)((base + row) * NBATCH + nIdx) * E_DIM + colg] =
            outAcc[r] / den;
      }
    }
  }
}

// ---------------------------------------------------------------------------
extern "C" void kernel_launch(void* const* d_in, const int* in_sizes, int n_in,
                              void* d_out, int out_size, void* d_ws,
                              size_t ws_size, hipStream_t stream) {
  (void)in_sizes; (void)n_in; (void)out_size; (void)ws_size;
  const float* query = (const float*)d_in[0];
  const float* q_w   = (const float*)d_in[1];
  const float* q_b   = (const float*)d_in[2];
  const float* k_w   = (const float*)d_in[3];
  const float* k_b   = (const float*)d_in[4];
  const float* v_w   = (const float*)d_in[5];
  const float* v_b   = (const float*)d_in[6];
  const float* o_w   = (const float*)d_in[7];
  const float* o_b   = (const float*)d_in[8];
  float* out = (float*)d_out;

  const size_t PLANE = (size_t)L_SEQ * NBATCH * E_DIM;   // 1 Mi floats
  float* qbuf = (float*)d_ws;
  float* kbuf = qbuf + PLANE;
  float* vbuf = kbuf + PLANE;
  float* abuf = vbuf + PLANE;

  dim3 gridG(E_DIM / 64, (L_SEQ * NBATCH) / 16);   // (8, 128)
  cosformer_proj_gemm<<<gridG, 128, 0, stream>>>(query, q_w, q_b, qbuf, 1);
  cosformer_proj_gemm<<<gridG, 128, 0, stream>>>(query, k_w, k_b, kbuf, 1);
  cosformer_proj_gemm<<<gridG, 128, 0, stream>>>(query, v_w, v_b, vbuf, 0);
  cosformer_scan<<<NH, 160, 0, stream>>>(qbuf, kbuf, vbuf, abuf);
  cosformer_proj_gemm<<<gridG, 128, 0, stream>>>(abuf, o_w, o_b, out, 0);
}